// multiscale_RNN_cell_90494960926876
// MI455X (gfx1250) — compile-verified
//
#include <hip/hip_runtime.h>

#define HIDDEN 48
#define INPUT  24
#define BATCH  524288
#define Z_LOW  0.005f
#define Z_HIGH 1.0f

typedef __attribute__((ext_vector_type(2))) float v2f;
typedef __attribute__((ext_vector_type(8))) float v8f;

__device__ __forceinline__ float sigmoidf(float v) {
    return 1.0f / (1.0f + __expf(-v));
}

// Each 256-thread block = 8 waves; each wave computes a 16-column batch tile.
// Stacked weights [W ; W_z] (96x48) and [Pm ; P_z] (96x24) live in LDS.
__global__ __launch_bounds__(256)
void multiscale_rnn_cell_kernel(const float* __restrict__ x,    // (INPUT, BATCH)
                                const float* __restrict__ r_t,  // (BATCH, HIDDEN)
                                const float* __restrict__ W,    // (H, H)
                                const float* __restrict__ P,    // (H, INPUT)
                                const float* __restrict__ b_v,  // (H)
                                const float* __restrict__ b_z,  // (H)
                                const float* __restrict__ W_z,  // (H, H)
                                const float* __restrict__ P_z,  // (H, INPUT)
                                float* __restrict__ out)        // (BATCH, HIDDEN)
{
    __shared__ float ldsW[2 * HIDDEN * HIDDEN];   // [W ; W_z], row-major, stride 48
    __shared__ float ldsP[2 * HIDDEN * INPUT];    // [Pm ; P_z], row-major, stride 24
    __shared__ float ldsB[2 * HIDDEN];            // [b_v ; b_z]
    __shared__ float stage[8 * 16 * HIDDEN];      // per-wave output staging (n,h)

    const int tid = threadIdx.x;

    // ---- stage weights into LDS (coalesced) ----
    for (int i = tid; i < HIDDEN * HIDDEN; i += 256) {
        ldsW[i] = W[i];
        ldsW[HIDDEN * HIDDEN + i] = W_z[i];
    }
    for (int i = tid; i < HIDDEN * INPUT; i += 256) {
        const int row = i / INPUT;
        ldsP[i] = (row < (HIDDEN / 2)) ? P[i] : 0.0f;   // mask rows >= H/2
        ldsP[HIDDEN * INPUT + i] = P_z[i];
    }
    if (tid < HIDDEN) {
        ldsB[tid] = b_v[tid];
        ldsB[HIDDEN + tid] = b_z[tid];
    }
    __syncthreads();

    const int wave = tid >> 5;
    const int lane = tid & 31;
    const int n16  = lane & 15;          // B/C/D column within tile
    const int kb   = (lane >> 4) << 1;   // A/B k sub-offset: 0 or 2
    const int hoff = (lane >> 4) << 3;   // C/D row offset: 0 or 8
    const long b0  = (long)blockIdx.x * 128 + (long)wave * 16;

    const v8f vzero = {0.f, 0.f, 0.f, 0.f, 0.f, 0.f, 0.f, 0.f};
    v8f accV[3], accZ[3];
#pragma unroll
    for (int t = 0; t < 3; ++t) { accV[t] = vzero; accZ[t] = vzero; }

    // ---- recurrent term: [W;W_z] @ r, K = 48 in steps of 4 ----
    const float* rrow = r_t + (b0 + n16) * HIDDEN + kb;   // per-lane row base
#pragma unroll
    for (int kk = 0; kk < HIDDEN / 4; ++kk) {
        const v2f bfrag = *(const v2f*)(rrow + kk * 4);   // global_load_b64
#pragma unroll
        for (int t = 0; t < 3; ++t) {
            const v2f aV = *(const v2f*)(&ldsW[(t * 16 + n16) * HIDDEN + kk * 4 + kb]);
            const v2f aZ = *(const v2f*)(&ldsW[HIDDEN * HIDDEN + (t * 16 + n16) * HIDDEN + kk * 4 + kb]);
            accV[t] = __builtin_amdgcn_wmma_f32_16x16x4_f32(
                false, aV, false, bfrag, (short)0, accV[t], false, false);
            accZ[t] = __builtin_amdgcn_wmma_f32_16x16x4_f32(
                false, aZ, false, bfrag, (short)0, accZ[t], false, false);
        }
    }

    // ---- input term: [Pm;P_z] @ x, K = 24 in steps of 4 ----
    const float* xcol = x + (long)kb * BATCH + b0 + n16;
#pragma unroll
    for (int kk = 0; kk < INPUT / 4; ++kk) {
        const float* xp = xcol + (long)(kk * 4) * BATCH;
        v2f bfrag;
        bfrag.x = xp[0];          // coalesced across 16-lane group
        bfrag.y = xp[BATCH];
#pragma unroll
        for (int t = 0; t < 3; ++t) {
            const v2f aV = *(const v2f*)(&ldsP[(t * 16 + n16) * INPUT + kk * 4 + kb]);
            const v2f aZ = *(const v2f*)(&ldsP[HIDDEN * INPUT + (t * 16 + n16) * INPUT + kk * 4 + kb]);
            accV[t] = __builtin_amdgcn_wmma_f32_16x16x4_f32(
                false, aV, false, bfrag, (short)0, accV[t], false, false);
            accZ[t] = __builtin_amdgcn_wmma_f32_16x16x4_f32(
                false, aZ, false, bfrag, (short)0, accZ[t], false, false);
        }
    }

    // ---- gate + blend (C/D layout: VGPR v -> rows v / v+8) ----
    float* st = &stage[wave * 16 * HIDDEN];
#pragma unroll
    for (int t = 0; t < 3; ++t) {
#pragma unroll
        for (int v = 0; v < 8; ++v) {
            const int h = t * 16 + hoff + v;
            const float sv = accV[t][v] + ldsB[h];
            const float sz = accZ[t][v] + ldsB[HIDDEN + h];
            const float z  = Z_LOW + (Z_HIGH - Z_LOW) * sigmoidf(sz);
            const float ro = r_t[(b0 + n16) * HIDDEN + h];   // cache hit (loaded above)
            st[n16 * HIDDEN + h] = (1.0f - z) * ro + z * sigmoidf(sv);
        }
    }
    __syncthreads();

    // ---- coalesced b128 store of the 16x48 tile ----
    const float* stw = &stage[wave * 16 * HIDDEN];
    float* op = out + b0 * HIDDEN;
#pragma unroll
    for (int i = 0; i < 6; ++i) {
        const int f = i * 128 + lane * 4;   // 768 floats total
        *(float4*)(op + f) = *(const float4*)(stw + f);
    }
}

extern "C" void kernel_launch(void* const* d_in, const int* in_sizes, int n_in,
                              void* d_out, int out_size, void* d_ws, size_t ws_size,
                              hipStream_t stream) {
    const float* x   = (const float*)d_in[0];
    const float* r_t = (const float*)d_in[1];
    const float* W   = (const float*)d_in[2];
    const float* P   = (const float*)d_in[3];
    const float* b_v = (const float*)d_in[4];
    const float* b_z = (const float*)d_in[5];
    const float* W_z = (const float*)d_in[6];
    const float* P_z = (const float*)d_in[7];
    float* out = (float*)d_out;

    dim3 grid(BATCH / 128);   // 4096 workgroups, 8 waves each, 16 cols/wave
    multiscale_rnn_cell_kernel<<<grid, 256, 0, stream>>>(
        x, r_t, W, P, b_v, b_z, W_z, P_z, out);
}